// Upsample_64948495450678
// MI455X (gfx1250) — compile-verified
//
#include <hip/hip_runtime.h>
#include <hip/hip_bf16.h>

// 2x FIR upsample (StyleGAN upfirdn2d, taps [1,3,3,1], factor 2).
// In : (8,64,256,256) f32   Out: (8,64,511,511) f32  (XLA lhs_dilation geometry)
// Memory-bound (~660 MB traffic): stage tiles via CDNA5 async global->LDS,
// compute 2x2 output quads per lane, non-temporal stores for the output.

#define IN_H   256
#define IN_W   256
#define OUT_H  511
#define OUT_W  511
#define NIMG   (8 * 64)

#define QTX    64              // quads per block, x
#define QTY    16              // quads per block, y
#define TILE_W (QTX + 1)       // 65
#define TILE_H (QTY + 1)       // 17
#define TILE_N (TILE_W * TILE_H) // 1105 floats = 4420 B LDS

__launch_bounds__(256, 8)
__global__ void upsample2x_fir_kernel(const float* __restrict__ x,
                                      float* __restrict__ out)
{
    __shared__ float tile[TILE_N];

    const int tid = threadIdx.x;
    const int img = blockIdx.z;
    const int qx0 = blockIdx.x * QTX;
    const int qy0 = blockIdx.y * QTY;

    const float* imgIn = x + (size_t)img * (IN_H * IN_W);

    // Lower 32 bits of a flat LDS pointer are the LDS byte address (aperture layout).
    const unsigned ldsBase = (unsigned)(uintptr_t)(&tile[0]);

    // ---- Stage (QTY+1) x (QTX+1) input tile into LDS -------------------------
    // Tile origin is (qy0-1, qx0-1); only the image top/left halo can be OOB
    // (high side is always in range by construction). OOB entries are zeroed
    // with a plain ds store; in-range entries use the CDNA5 async copy path
    // (global_load_async_to_lds_b32, tracked by ASYNCcnt).
    for (int l = tid; l < TILE_N; l += 256) {
        const int row = l / TILE_W;
        const int col = l - row * TILE_W;
        const int gy  = qy0 - 1 + row;
        const int gx  = qx0 - 1 + col;
        const unsigned ldsOff = ldsBase + (unsigned)l * 4u;
        if (gy >= 0 && gx >= 0) {
            const unsigned gOff = (unsigned)(gy * IN_W + gx) * 4u;
            asm volatile("global_load_async_to_lds_b32 %0, %1, %2"
                         :
                         : "v"(ldsOff), "v"(gOff), "s"(imgIn)
                         : "memory");
        } else {
            tile[l] = 0.0f;
        }
    }
    asm volatile("s_wait_asynccnt 0" ::: "memory");
    __syncthreads();

    // ---- Compute: each lane owns 4 output quads (one per iteration) ----------
    // Quad (qy,qx) -> outputs (2qy-1, 2qy) x (2qx-1, 2qx) from inputs
    // a=x[qy-1,qx-1] b=x[qy-1,qx] c=x[qy,qx-1] d=x[qy,qx] (zeros pre-staged).
    const int tx  = tid & (QTX - 1);   // 0..63: quad column (wave-contiguous)
    const int tyb = tid >> 6;          // 0..3
    float* imgOut = out + (size_t)img * (OUT_H * OUT_W);
    const float inv16 = 0.0625f;

#pragma unroll
    for (int j = 0; j < 4; ++j) {
        const int r = tyb * 4 + j;     // quad row 0..15
        const float* t0 = &tile[r * TILE_W + tx];
        const float a = t0[0];
        const float b = t0[1];
        const float c = t0[TILE_W];
        const float d = t0[TILE_W + 1];

        const float voo = (9.0f * a + 3.0f * b + 3.0f * c +        d) * inv16;
        const float voe = (3.0f * a + 9.0f * b +        c + 3.0f * d) * inv16;
        const float veo = (3.0f * a +        b + 9.0f * c + 3.0f * d) * inv16;
        const float vee = (       a + 3.0f * b + 3.0f * c + 9.0f * d) * inv16;

        const int qy  = qy0 + r;
        const int qx  = qx0 + tx;
        const int oyo = 2 * qy - 1;    // may be -1 at the image top
        const int oxo = 2 * qx - 1;    // may be -1 at the image left

        float* rowE = imgOut + (size_t)(2 * qy) * OUT_W;  // 2qy <= 510: in range
        if (oyo >= 0) {
            float* rowO = imgOut + (size_t)oyo * OUT_W;
            if (oxo >= 0) __builtin_nontemporal_store(voo, rowO + oxo);
            __builtin_nontemporal_store(voe, rowO + 2 * qx);
        }
        if (oxo >= 0) __builtin_nontemporal_store(veo, rowE + oxo);
        __builtin_nontemporal_store(vee, rowE + 2 * qx);
    }
}

extern "C" void kernel_launch(void* const* d_in, const int* in_sizes, int n_in,
                              void* d_out, int out_size, void* d_ws, size_t ws_size,
                              hipStream_t stream)
{
    (void)in_sizes; (void)n_in; (void)d_ws; (void)ws_size; (void)out_size;
    const float* x = (const float*)d_in[0];
    // d_in[1] is the fixed FIR taps [1,3,3,1]; weights are baked into the kernel.
    float* out = (float*)d_out;

    dim3 grid(IN_W / QTX, IN_H / QTY, NIMG);  // (4, 16, 512)
    upsample2x_fir_kernel<<<grid, 256, 0, stream>>>(x, out);
}